// coxloss_34445637714569
// MI455X (gfx1250) — compile-verified
//
#include <hip/hip_runtime.h>
#include <hip/hip_bf16.h>
#include <math.h>

// Problem constants (match reference)
#define E_TOTAL   8192
#define ROW_LEN   8192
#define N_PRED    100000

// Tuning: cache first NCACHE pred values in LDS (319488 bytes < 320KB WGP LDS)
#define NCACHE    79872
#define BLOCK     1024          // 32 wave32 waves per block
#define WAVES     (BLOCK / 32)
#define EPB       WAVES         // one event per wave
#define GRID      (E_TOTAL / EPB)   // 256 blocks

#define AS1 __attribute__((address_space(1)))
#define AS3 __attribute__((address_space(3)))

#if defined(__has_builtin)
#  if __has_builtin(__builtin_amdgcn_global_load_async_to_lds_b128) && \
      __has_builtin(__builtin_amdgcn_s_wait_asynccnt)
#    define USE_ASYNC_FILL 1
#  endif
#endif
#ifndef USE_ASYNC_FILL
#  define USE_ASYNC_FILL 0
#endif

typedef int v4i __attribute__((ext_vector_type(4)));
// Exact pointee type of the async-to-LDS builtin's pointer params
// (per compiler diagnostic: vector_size(16) int, AS1 src / AS3 dst).
typedef int v4i_b128 __attribute__((vector_size(16)));

__device__ __forceinline__ void online_update(float x, float& m, float& s) {
    // streaming logsumexp state (m, s) with a single exp:
    // t = exp(-|x-m|); if x>m: s = s*t + 1 else s = s + t; m = max(m,x)
    float t = __expf(-fabsf(x - m));
    bool  g = x > m;
    s = g ? __fmaf_rn(s, t, 1.0f) : (s + t);
    m = fmaxf(m, x);
}

__global__ __launch_bounds__(BLOCK)
void cox_partial_kernel(const float* __restrict__ pred,
                        const int*   __restrict__ risk_idx,
                        const int*   __restrict__ set_len,
                        float*       __restrict__ part) {
    __shared__ float cache[NCACHE];

    // Fill the LDS pred-cache (source is L2-resident after first touch).
#if USE_ASYNC_FILL
    // CDNA5 async copy: global -> LDS directly, no VGPR round trip, no
    // per-iteration loadcnt stall. Tracked by ASYNCcnt.
    for (int i = threadIdx.x * 4; i < NCACHE; i += BLOCK * 4) {
        __builtin_amdgcn_global_load_async_to_lds_b128(
            (AS1 v4i_b128*)(pred + i),
            (AS3 v4i_b128*)(cache + i),
            0, 0);
    }
    __builtin_amdgcn_s_wait_asynccnt(0);
#else
    {
        const float4* __restrict__ p4 = reinterpret_cast<const float4*>(pred);
        float4* __restrict__ c4 = reinterpret_cast<float4*>(cache);
        for (int i = threadIdx.x; i < NCACHE / 4; i += BLOCK)
            c4[i] = p4[i];
    }
#endif
    __syncthreads();

    const int lane = threadIdx.x & 31;
    const int wave = threadIdx.x >> 5;
    const int e    = blockIdx.x * EPB + wave;   // grid covers exactly E_TOTAL

    const int n_valid = max(set_len[e], 1);
    const int* __restrict__ row = risk_idx + (size_t)e * ROW_LEN;
    const v4i* __restrict__ rowv = reinterpret_cast<const v4i*>(row);

    float m = -3.0e38f;   // finite sentinel: exp underflows to 0, no NaN paths
    float s = 0.0f;

    const int nchunk = (n_valid + 3) >> 2;   // int4 chunks covering valid region
    for (int c = lane; c < nchunk; c += 32) {
        // Streaming, read-once index data: non-temporal so pred stays hot in L2.
        v4i iv = __builtin_nontemporal_load(rowv + c);
        int base = c << 2;
        if (base + 3 < n_valid) {
            #pragma unroll
            for (int k = 0; k < 4; ++k) {
                int idx = iv[k];
                float x = (idx < NCACHE) ? cache[idx] : pred[idx];
                online_update(x, m, s);
            }
        } else {
            #pragma unroll
            for (int k = 0; k < 4; ++k) {
                if (base + k < n_valid) {
                    int idx = iv[k];
                    float x = (idx < NCACHE) ? cache[idx] : pred[idx];
                    online_update(x, m, s);
                }
            }
        }
    }

    // Wave32 butterfly reduction of (m, s) pairs.
    #pragma unroll
    for (int off = 16; off > 0; off >>= 1) {
        float mo = __shfl_xor(m, off, 32);
        float so = __shfl_xor(s, off, 32);
        float M  = fmaxf(m, mo);
        s = s * __expf(m - M) + so * __expf(mo - M);
        m = M;
    }

    if (lane == 0) {
        float s0 = pred[row[0]];               // column 0 is always valid
        part[e] = (m + __logf(s)) - s0;        // lse - score0
    }
}

__global__ __launch_bounds__(256)
void cox_reduce_kernel(const float* __restrict__ part, float* __restrict__ out) {
    __shared__ float red[256 / 32];
    float acc = 0.0f;
    for (int i = threadIdx.x; i < E_TOTAL; i += 256)
        acc += part[i];
    #pragma unroll
    for (int off = 16; off > 0; off >>= 1)
        acc += __shfl_xor(acc, off, 32);
    const int lane = threadIdx.x & 31;
    const int wave = threadIdx.x >> 5;
    if (lane == 0) red[wave] = acc;
    __syncthreads();
    if (threadIdx.x == 0) {
        float t = 0.0f;
        #pragma unroll
        for (int w = 0; w < 256 / 32; ++w) t += red[w];
        out[0] = t;   // loss = sum(lse - s0) = -z
    }
}

extern "C" void kernel_launch(void* const* d_in, const int* in_sizes, int n_in,
                              void* d_out, int out_size, void* d_ws, size_t ws_size,
                              hipStream_t stream) {
    const float* pred     = (const float*)d_in[0];
    const int*   risk_idx = (const int*)d_in[1];
    const int*   set_len  = (const int*)d_in[2];
    float*       out      = (float*)d_out;
    float*       part     = (float*)d_ws;     // E_TOTAL floats of scratch

    cox_partial_kernel<<<GRID, BLOCK, 0, stream>>>(pred, risk_idx, set_len, part);
    cox_reduce_kernel<<<1, 256, 0, stream>>>(part, out);
}